// GCN_23124103922300
// MI455X (gfx1250) — compile-verified
//
#include <hip/hip_runtime.h>

#define HDIM 128
#define ODIM 32
#define EPSL 1e-5f

typedef __attribute__((ext_vector_type(2))) float v2f;
typedef __attribute__((ext_vector_type(8))) float v8f;

// ------------------------------------------------------------------
// degree / normalization
// ------------------------------------------------------------------
__global__ void k_init_deg(float* __restrict__ deg, int n) {
    int i = blockIdx.x * blockDim.x + threadIdx.x;
    if (i < n) deg[i] = 1.0f;   // self-loop contributes 1
}

__global__ void k_deg_edges(const int* __restrict__ dst, float* __restrict__ deg, int e) {
    int i = blockIdx.x * blockDim.x + threadIdx.x;
    if (i < e) atomicAdd(&deg[dst[i]], 1.0f);
}

__global__ void k_dis(float* __restrict__ deg, int n) {
    int i = blockIdx.x * blockDim.x + threadIdx.x;
    if (i < n) deg[i] = rsqrtf(deg[i]);   // deg >= 1 always (self-loops)
}

// ------------------------------------------------------------------
// node embedding: h[v] = class_emb[x[v,0]] + neigh_emb[x[v,1]]
// one wave per node, float4 per lane (32*4 = 128)
// ------------------------------------------------------------------
__global__ __launch_bounds__(256) void k_embed(const int* __restrict__ x,
                                               const float* __restrict__ ce,
                                               const float* __restrict__ ne,
                                               float* __restrict__ h, int n) {
    int wid  = (blockIdx.x * blockDim.x + threadIdx.x) >> 5;
    int lane = threadIdx.x & 31;
    if (wid >= n) return;
    int c0 = x[wid * 2 + 0];
    int c1 = x[wid * 2 + 1];
    float4 a = ((const float4*)(ce + (size_t)c0 * HDIM))[lane];
    float4 b = ((const float4*)(ne + (size_t)c1 * HDIM))[lane];
    float4 r = make_float4(a.x + b.x, a.y + b.y, a.z + b.z, a.w + b.w);
    ((float4*)(h + (size_t)wid * HDIM))[lane] = r;
}

// ------------------------------------------------------------------
// hw = h @ W   (N x 128) @ (128 x 128), f32 WMMA 16x16x4
// block: 256 threads = 8 waves, covers 64 rows x 128 cols.
// wave w owns column panel [16w, 16w+16); computes 4 M-tiles of 16 rows.
// A tile (64x128) staged in LDS padded to 130 floats/row (conflict-free:
// bank = (row*2 + col) % 64 distinct across the 16 M-lanes).
//
// f32 WMMA operand layout (ISA 7.12.2):
//   A 16x4 : lane = M (0..15 twice); lane<16 -> K={0,1}, lane>=16 -> K={2,3};
//            VGPR0=K even, VGPR1=K odd  => contiguous float2 per lane.
//   B 4x16 : lane = N; lane-half selects K rows {0,1} / {2,3} (mirror of A).
//   C/D    : VGPR r, lane l -> row = r + 8*(l>>4), col = l&15.
// ------------------------------------------------------------------
__global__ __launch_bounds__(256) void k_gemm(const float* __restrict__ A,
                                              const float* __restrict__ W,
                                              float* __restrict__ Out, int n) {
    __shared__ float sA[64 * 130];
    const int tid  = threadIdx.x;
    const int lane = tid & 31;
    const int wave = tid >> 5;
    const size_t row0 = (size_t)blockIdx.x * 64;

    // stage 64x128 slab of A into padded LDS
    for (int idx = tid; idx < 64 * HDIM; idx += 256) {
        int r = idx >> 7, c = idx & 127;
        sA[r * 130 + c] = A[(row0 + r) * HDIM + c];
    }
    __syncthreads();

    const int nb = wave * 16;     // column base of this wave's panel
    const int lm = lane & 15;     // M (for A) / N (for B,C)
    const int kh = lane >> 4;     // K half selector

    v8f acc[4] = {};              // 4 M-tiles x 8 VGPRs
    for (int k0 = 0; k0 < HDIM; k0 += 4) {
        const int kk = k0 + 2 * kh;
        v2f b;
        b.x = W[kk * HDIM + nb + lm];
        b.y = W[(kk + 1) * HDIM + nb + lm];
#pragma unroll
        for (int m = 0; m < 4; ++m) {
            const v2f a = *(const v2f*)&sA[(m * 16 + lm) * 130 + kk];
            acc[m] = __builtin_amdgcn_wmma_f32_16x16x4_f32(
                false, a, false, b, (short)0, acc[m], false, false);
        }
    }

#pragma unroll
    for (int m = 0; m < 4; ++m) {
#pragma unroll
        for (int r = 0; r < 8; ++r) {
            size_t row = row0 + (size_t)(m * 16 + r + 8 * kh);
            Out[row * HDIM + nb + lm] = acc[m][r];
        }
    }
}

// ------------------------------------------------------------------
// agg[v] = bias + dis[v]^2 * hw[v]     (self-loop term + bias)
// ------------------------------------------------------------------
__global__ __launch_bounds__(256) void k_init_agg(const float* __restrict__ hw,
                                                  const float* __restrict__ dis,
                                                  const float* __restrict__ bias,
                                                  float* __restrict__ agg, int n) {
    int wid  = (blockIdx.x * blockDim.x + threadIdx.x) >> 5;
    int lane = threadIdx.x & 31;
    if (wid >= n) return;
    float d  = dis[wid];
    float dd = d * d;
    float4 v = ((const float4*)(hw + (size_t)wid * HDIM))[lane];
    float4 b = ((const float4*)bias)[lane];
    float4 r = make_float4(b.x + dd * v.x, b.y + dd * v.y,
                           b.z + dd * v.z, b.w + dd * v.w);
    ((float4*)(agg + (size_t)wid * HDIM))[lane] = r;
}

// ------------------------------------------------------------------
// edge scatter: agg[dst] += dis[src]*dis[dst] * hw[src]
// one wave per edge; 786K x 512B stays hot in the 192MB L2
// ------------------------------------------------------------------
__global__ __launch_bounds__(256) void k_scatter(const float* __restrict__ hw,
                                                 const int* __restrict__ src,
                                                 const int* __restrict__ dst,
                                                 const float* __restrict__ dis,
                                                 float* __restrict__ agg, int e) {
    int wid  = (blockIdx.x * blockDim.x + threadIdx.x) >> 5;
    int lane = threadIdx.x & 31;
    if (wid >= e) return;
    int s = src[wid];
    int d = dst[wid];
    float nrm = dis[s] * dis[d];
    float4 v  = ((const float4*)(hw + (size_t)s * HDIM))[lane];
    float* pd = agg + (size_t)d * HDIM + lane * 4;
    atomicAdd(pd + 0, nrm * v.x);
    atomicAdd(pd + 1, nrm * v.y);
    atomicAdd(pd + 2, nrm * v.z);
    atomicAdd(pd + 3, nrm * v.w);
}

// ------------------------------------------------------------------
// h = LayerNorm(h + relu(agg)) * g + b   — wave-per-node, shfl_xor reductions
// ------------------------------------------------------------------
__global__ __launch_bounds__(256) void k_update(float* __restrict__ h,
                                                const float* __restrict__ agg,
                                                const float* __restrict__ g,
                                                const float* __restrict__ b, int n) {
    int wid  = (blockIdx.x * blockDim.x + threadIdx.x) >> 5;
    int lane = threadIdx.x & 31;
    if (wid >= n) return;
    float4 hv = ((const float4*)(h + (size_t)wid * HDIM))[lane];
    float4 av = ((const float4*)(agg + (size_t)wid * HDIM))[lane];
    float4 t  = make_float4(hv.x + fmaxf(av.x, 0.0f),
                            hv.y + fmaxf(av.y, 0.0f),
                            hv.z + fmaxf(av.z, 0.0f),
                            hv.w + fmaxf(av.w, 0.0f));
    float s = t.x + t.y + t.z + t.w;
#pragma unroll
    for (int o = 16; o >= 1; o >>= 1) s += __shfl_xor(s, o, 32);
    float mu = s * (1.0f / HDIM);
    float4 c = make_float4(t.x - mu, t.y - mu, t.z - mu, t.w - mu);
    float q = c.x * c.x + c.y * c.y + c.z * c.z + c.w * c.w;
#pragma unroll
    for (int o = 16; o >= 1; o >>= 1) q += __shfl_xor(q, o, 32);
    float r = rsqrtf(q * (1.0f / HDIM) + EPSL);
    float4 gg = ((const float4*)g)[lane];
    float4 bb = ((const float4*)b)[lane];
    float4 out = make_float4(c.x * r * gg.x + bb.x, c.y * r * gg.y + bb.y,
                             c.z * r * gg.z + bb.z, c.w * r * gg.w + bb.w);
    ((float4*)(h + (size_t)wid * HDIM))[lane] = out;
}

// ------------------------------------------------------------------
// out[t] = h[root_t] @ out_w   (256 x 128) @ (128 x 32) — tiny
// ------------------------------------------------------------------
__global__ void k_out(const float* __restrict__ h, const float* __restrict__ ow,
                      float* __restrict__ out, int nodes_per_tree) {
    int t = blockIdx.x;
    int o = threadIdx.x;          // 32 threads
    const float* hr = h + (size_t)t * nodes_per_tree * HDIM;
    float acc = 0.0f;
    for (int k = 0; k < HDIM; ++k) acc += hr[k] * ow[k * ODIM + o];
    out[t * ODIM + o] = acc;
}

// ------------------------------------------------------------------
extern "C" void kernel_launch(void* const* d_in, const int* in_sizes, int n_in,
                              void* d_out, int out_size, void* d_ws, size_t ws_size,
                              hipStream_t stream) {
    const int*   x    = (const int*)d_in[0];                  // [N,2]
    const int*   ei   = (const int*)d_in[1];                  // [2,E]
    const float* ce   = (const float*)d_in[3];                // [32,128]
    const float* ne   = (const float*)d_in[4];                // [33,128]
    const float* Ws   = (const float*)d_in[5];                // [4,128,128]
    const float* bs   = (const float*)d_in[6];                // [4,128]
    const float* lng  = (const float*)d_in[7];                // [4,128]
    const float* lnb  = (const float*)d_in[8];                // [4,128]
    const float* outw = (const float*)d_in[9];                // [128,32]
    float* out = (float*)d_out;

    const int n = in_sizes[0] / 2;
    const int e = in_sizes[1] / 2;
    const int L = in_sizes[6] / HDIM;          // 4
    const int ntrees = out_size / ODIM;        // 256
    const int nodes_per_tree = n / ntrees;

    const int* src = ei;
    const int* dst = ei + e;

    float* h   = (float*)d_ws;
    float* hw  = h  + (size_t)n * HDIM;
    float* agg = hw + (size_t)n * HDIM;
    float* dis = agg + (size_t)n * HDIM;

    const int TB = 256;
    const int wavesPerBlk = TB / 32;
    const int nodeBlks = (n + wavesPerBlk - 1) / wavesPerBlk;
    const int edgeBlks = (e + wavesPerBlk - 1) / wavesPerBlk;

    // normalization: deg -> dis
    k_init_deg<<<(n + TB - 1) / TB, TB, 0, stream>>>(dis, n);
    k_deg_edges<<<(e + TB - 1) / TB, TB, 0, stream>>>(dst, dis, e);
    k_dis<<<(n + TB - 1) / TB, TB, 0, stream>>>(dis, n);

    // embedding
    k_embed<<<nodeBlks, TB, 0, stream>>>(x, ce, ne, h, n);

    // layers
    for (int i = 0; i < L; ++i) {
        k_gemm<<<(n + 63) / 64, TB, 0, stream>>>(h, Ws + (size_t)i * HDIM * HDIM, hw, n);
        k_init_agg<<<nodeBlks, TB, 0, stream>>>(hw, dis, bs + (size_t)i * HDIM, agg, n);
        k_scatter<<<edgeBlks, TB, 0, stream>>>(hw, src, dst, dis, agg, e);
        k_update<<<nodeBlks, TB, 0, stream>>>(h, agg, lng + (size_t)i * HDIM,
                                              lnb + (size_t)i * HDIM, n);
    }

    // readout
    k_out<<<ntrees, ODIM, 0, stream>>>(h, outw, out, nodes_per_tree);
}